// FCOSLayer_7696581394898
// MI455X (gfx1250) — compile-verified
//
#include <hip/hip_runtime.h>
#include <math.h>

#define NB     16
#define NGT    50
#define NG     64
#define NA     3
#define NCH    85
#define NCLS   80
#define STRIDEF 8.0f

typedef unsigned int u32x4 __attribute__((ext_vector_type(4)));
typedef int          i32x8 __attribute__((ext_vector_type(8)));
typedef int          i32x4 __attribute__((ext_vector_type(4)));

__device__ __forceinline__ float bce_logits(float x, float t) {
    return fmaxf(x, 0.f) - x * t + log1pf(expf(-fabsf(x)));
}

// ---------------------------------------------------------------------------
// TDM 2D tile load: dim0 contiguous elements x dim1 lines (stride elements
// apart) -> LDS. D# built per CDNA5 ISA ch.8 (groups 0/1; groups 2/3 zero,
// dims >=2 unused). data_size = 4 bytes. Issued once per calling wave.
// 6-arg builtin form (amdgpu-toolchain / therock-10.0 signature).
// ---------------------------------------------------------------------------
__device__ __forceinline__ void tdm_load_2d_f32(unsigned lds_byte_addr,
                                                const void* gaddr,
                                                unsigned dim0_elems,
                                                unsigned dim1_lines,
                                                unsigned stride_elems) {
    unsigned long long ga = (unsigned long long)gaddr;
    u32x4 g0;
    g0[0] = 1u;                                            // count=1 (valid), user mode
    g0[1] = lds_byte_addr;                                 // lds_addr
    g0[2] = (unsigned)(ga & 0xFFFFFFFFull);                // global_addr[31:0]
    g0[3] = (unsigned)((ga >> 32) & 0x1FFFFFFull)          // global_addr[56:32]
          | (2u << 30);                                    // type=2 ("image")
    i32x8 g1;
    g1[0] = (int)(2u << 16);                               // data_size=2 (4B), wg_mask=0
    g1[1] = (int)((dim0_elems & 0xFFFFu) << 16);           // tensor_dim0[15:0]
    g1[2] = (int)(((dim0_elems >> 16) & 0xFFFFu)           // tensor_dim0[31:16]
          | ((dim1_lines & 0xFFFFu) << 16));               // tensor_dim1[15:0]
    g1[3] = (int)(((dim1_lines >> 16) & 0xFFFFu)           // tensor_dim1[31:16]
          | ((dim0_elems & 0xFFFFu) << 16));               // tile_dim0 (16b)
    g1[4] = (int)(dim1_lines & 0xFFFFu);                   // tile_dim1; tile_dim2=0
    g1[5] = (int)stride_elems;                             // tensor_dim0_stride[31:0]
    g1[6] = 0;                                             // stride[47:32]=0, dim1_stride lo=0
    g1[7] = 0;                                             // tensor_dim1_stride rest
    i32x4 z4 = {0, 0, 0, 0};
    i32x8 z8 = {0, 0, 0, 0, 0, 0, 0, 0};
    __builtin_amdgcn_tensor_load_to_lds(g0, g1, z4, z4, z8, 0);
}

// ---------------------------------------------------------------------------
// Kernel 1: per-GT preprocessing. grid=NB, block=64 (2 wave32 waves).
// rec layout: gx gy gw gh valid best_n ti tj tl0 tl1 tl2 tl3 w cls pad pad
// ---------------------------------------------------------------------------
__global__ void gt_prep_kernel(const float* __restrict__ labels,
                               const float* __restrict__ all_anchors,
                               const int*   __restrict__ anchor_indices,
                               const int*   __restrict__ img_size,
                               float* __restrict__ rec,
                               int*   __restrict__ anyv,
                               float* __restrict__ out) {
    int b = blockIdx.x;
    int g = threadIdx.x;
    __shared__ int s_any;
    if (g == 0) s_any = 0;
    if (b == 0 && g == 0) out[0] = 0.f;   // kernels 2/3 accumulate atomically later
    __syncthreads();
    if (g < NGT) {
        const float* L = labels + (b * NGT + g) * 5;
        float cls = L[0];
        float gx = L[1], gy = L[2], gw = L[3], gh = L[4];
        int bestk = 0; float bestv = -1.f;            // first-max argmax over 9 anchors
        for (int k = 0; k < 9; ++k) {
            float aw = all_anchors[2*k], ah = all_anchors[2*k+1];
            float inter = fminf(gw, aw) * fminf(gh, ah);
            float iou = inter / (gw*gh + aw*ah - inter);
            if (iou > bestv) { bestv = iou; bestk = k; }
        }
        int best_n = bestk % NA;
        int valid = 0;
        for (int m = 0; m < NA; ++m) if (bestk == anchor_indices[m]) valid = 1;
        if (valid) atomicOr(&s_any, 1);
        float fimg = (float)img_size[0];
        int ti = (int)(gx / STRIDEF);
        int tj = (int)(gy / STRIDEF);
        float ccx = (floorf(gx / STRIDEF) + 0.5f) * STRIDEF;
        float ccy = (floorf(gy / STRIDEF) + 0.5f) * STRIDEF;
        float l   = fmaxf(ccx - (gx - gw*0.5f), 0.f);
        float t   = fmaxf(ccy - (gy - gh*0.5f), 0.f);
        float r   = fmaxf(gx + gw*0.5f - ccx, 0.f);
        float btm = fmaxf(gy + gh*0.5f - ccy, 0.f);
        float aw = all_anchors[2*anchor_indices[best_n]];
        float ah = all_anchors[2*anchor_indices[best_n]+1];
        float* R = rec + (b * NGT + g) * 16;
        R[0]=gx; R[1]=gy; R[2]=gw; R[3]=gh;
        R[4]=(float)valid; R[5]=(float)best_n; R[6]=(float)ti; R[7]=(float)tj;
        R[8]  = logf(l   / aw + 1e-8f);
        R[9]  = logf(t   / ah + 1e-8f);
        R[10] = logf(r   / aw + 1e-8f);
        R[11] = logf(btm / ah + 1e-8f);
        R[12] = 2.f - gw*gh/(fimg*fimg);
        R[13] = cls; R[14]=0.f; R[15]=0.f;
    }
    __syncthreads();
    if (g == 0) anyv[b] = s_any;
}

// ---------------------------------------------------------------------------
// Kernel 2: fused pred-decode + 50-GT IoU max + conf BCE.
// grid = NB*NA*4 (16 rows of one (b,a) plane per block), block = 256 (8 waves).
// - 5-channel raw tile (5 x 1024 f32, stride 4096) DMA'd to LDS via TDM
//   (tensor_load_to_lds, TENSORcnt), issued by wave 0.
// - 50-GT record table (800 dwords) staged via async global->LDS copies
//   (ASYNCcnt), issued by all waves. Both overlap, then one barrier.
// ---------------------------------------------------------------------------
__global__ void conf_loss_kernel(const float* __restrict__ raw,
                                 const float* __restrict__ all_anchors,
                                 const int*   __restrict__ anchor_indices,
                                 const int*   __restrict__ img_size,
                                 const float* __restrict__ rec,
                                 const int*   __restrict__ anyv,
                                 float* __restrict__ out) {
    __shared__ float sraw[5 * 1024];  // 20 KB: channels 0..4, 16-row chunk
    __shared__ float sgt[NGT * 16];   // 3.2 KB GT table
    __shared__ float sred[256];
    int blk    = blockIdx.x;
    int jchunk = blk & 3;
    int a      = (blk >> 2) % NA;
    int b      = blk / (4 * NA);
    int tid    = threadIdx.x;

    const float* rbase = raw + ((size_t)b * (NA*NCH) + (size_t)a * NCH) * (NG*NG);

    // --- TDM: 2D tile {1024 x 5ch, stride 4096} -> LDS (wave 0 only) ---
    if (tid == 0) {
        unsigned lds = (unsigned)(unsigned long long)(&sraw[0]);
        tdm_load_2d_f32(lds, rbase + jchunk * 16 * NG, 1024u, 5u, (unsigned)(NG*NG));
    }
    // --- async global->LDS staging of GT records (all waves) ---
    const float* src = rec + b * NGT * 16;
    for (int idx = tid; idx < NGT * 16; idx += 256) {
        unsigned lds = (unsigned)(unsigned long long)(&sgt[idx]);
        unsigned long long ga = (unsigned long long)(src + idx);
        asm volatile("global_load_async_to_lds_b32 %0, %1, off"
                     :: "v"(lds), "v"(ga) : "memory");
    }
    asm volatile("s_wait_asynccnt 0" ::: "memory");
    if (tid == 0) __builtin_amdgcn_s_wait_tensorcnt(0);
    __syncthreads();

    float fimg = (float)img_size[0];
    float aw = all_anchors[2*anchor_indices[a]];
    float ah = all_anchors[2*anchor_indices[a]+1];
    int anyvb = anyv[b];

    float lsum = 0.f;
    for (int it = 0; it < 4; ++it) {
        int n = tid + it * 256;            // index into the 1024-cell LDS tile
        int i = n & (NG - 1);
        int j = jchunk * 16 + (n >> 6);
        float r0 = sraw[n];
        float r1 = sraw[n + 1024];
        float r2 = sraw[n + 2048];
        float r3 = sraw[n + 3072];
        float x4 = sraw[n + 4096];
        float l0 = fminf(expf(r0) * aw, fimg);
        float l1 = fminf(expf(r1) * ah, fimg);
        float l2 = fminf(expf(r2) * aw, fimg);
        float l3 = fminf(expf(r3) * ah, fimg);
        float px = ((float)i + 0.5f) * STRIDEF + (l2 - l0) * 0.5f;
        float py = ((float)j + 0.5f) * STRIDEF + (l3 - l1) * 0.5f;
        float pw = l0 + l2;
        float ph = l1 + l3;
        float pxh = pw * 0.5f, pyh = ph * 0.5f;
        float area_p = pw * ph;
        float ioumax = 0.f;
        bool pos = false;
        for (int g = 0; g < NGT; ++g) {
            const float* G = &sgt[g * 16];
            float gx = G[0], gy = G[1], gw = G[2], gh = G[3];
            float tlx = fmaxf(px - pxh, gx - gw * 0.5f);
            float tly = fmaxf(py - pyh, gy - gh * 0.5f);
            float brx = fminf(px + pxh, gx + gw * 0.5f);
            float bry = fminf(py + pyh, gy + gh * 0.5f);
            float iw = fmaxf(brx - tlx, 0.f);
            float ih = fmaxf(bry - tly, 0.f);
            float inter = iw * ih;
            float iou = inter / (area_p + gw * gh - inter + 1e-12f);
            ioumax = fmaxf(ioumax, iou);
            if (G[4] != 0.f && (int)G[5] == a && (int)G[6] == i && (int)G[7] == j)
                pos = true;
        }
        bool mask = anyvb ? (ioumax < 0.6f) : true;
        if (pos) mask = true;
        float t = pos ? 1.f : 0.f;
        if (mask) lsum += bce_logits(x4, t);
    }
    sred[tid] = lsum;
    __syncthreads();
    for (int s = 128; s > 0; s >>= 1) {
        if (tid < s) sred[tid] += sred[tid + s];
        __syncthreads();
    }
    if (tid == 0) atomicAdd(out, sred[0]);
}

// ---------------------------------------------------------------------------
// Kernel 3: bbox + cls loss at positive cells only (sparse channel reads).
// Duplicate scatter targets resolved as "last valid writer wins".
// grid = NB, block = 64.
// ---------------------------------------------------------------------------
__global__ void pos_loss_kernel(const float* __restrict__ raw,
                                const float* __restrict__ rec,
                                float* __restrict__ out) {
    int b = blockIdx.x;
    int g = threadIdx.x;
    __shared__ float sred[64];
    float lsum = 0.f;
    if (g < NGT) {
        const float* R = rec + (b * NGT + g) * 16;
        if (R[4] != 0.f) {
            int a = (int)R[5], ti = (int)R[6], tj = (int)R[7];
            const float* rb = raw +
                (((size_t)b * (NA*NCH) + (size_t)a * NCH) * NG + tj) * NG + ti;
            __builtin_prefetch(rb, 0, 0);   // global_prefetch_b8 (speculative)
            bool winner = true;
            for (int g2 = g + 1; g2 < NGT; ++g2) {
                const float* R2 = rec + (b * NGT + g2) * 16;
                if (R2[4] != 0.f && (int)R2[5] == a && (int)R2[6] == ti && (int)R2[7] == tj) {
                    winner = false; break;
                }
            }
            if (winner) {
                float w = R[12];
                for (int c = 0; c < 4; ++c) {
                    float d = rb[(size_t)c * NG*NG] - R[8 + c];
                    lsum += 0.5f * w * d * d;
                }
                int cls = (int)R[13];
                for (int c = 0; c < NCLS; ++c) {
                    float x = rb[(size_t)(5 + c) * NG*NG];
                    lsum += bce_logits(x, (c == cls) ? 1.f : 0.f);
                }
            }
        }
    }
    sred[g] = lsum;
    __syncthreads();
    for (int s = 32; s > 0; s >>= 1) {
        if (g < s) sred[g] += sred[g + s];
        __syncthreads();
    }
    if (g == 0) atomicAdd(out, sred[0]);
}

extern "C" void kernel_launch(void* const* d_in, const int* in_sizes, int n_in,
                              void* d_out, int out_size, void* d_ws, size_t ws_size,
                              hipStream_t stream) {
    const float* raw            = (const float*)d_in[0];
    const float* labels         = (const float*)d_in[1];
    const float* all_anchors    = (const float*)d_in[2];
    const int*   anchor_indices = (const int*)d_in[3];
    const int*   img_size       = (const int*)d_in[4];
    float* out = (float*)d_out;

    float* rec  = (float*)d_ws;                                    // NB*NGT*16 floats
    int*   anyv = (int*)((char*)d_ws + NB * NGT * 16 * sizeof(float));

    gt_prep_kernel<<<NB, 64, 0, stream>>>(labels, all_anchors, anchor_indices,
                                          img_size, rec, anyv, out);
    conf_loss_kernel<<<NB * NA * 4, 256, 0, stream>>>(raw, all_anchors, anchor_indices,
                                                      img_size, rec, anyv, out);
    pos_loss_kernel<<<NB, 64, 0, stream>>>(raw, rec, out);
}